// AttentionProb_36876589204229
// MI455X (gfx1250) — compile-verified
//
#include <hip/hip_runtime.h>

typedef float  v2f   __attribute__((ext_vector_type(2)));
typedef float  v4f   __attribute__((ext_vector_type(4)));
typedef float  v8f   __attribute__((ext_vector_type(8)));
typedef long long v4i64 __attribute__((ext_vector_type(4)));

#define NNODES 100000
#define DIM    128
#define NEDGES 1600000

// ---------------------------------------------------------------------------
// Phase 1: scores = X (100000x128, fp32) @ [w_src | w_dst] via
// V_WMMA_F32_16X16X4_F32. One wave computes 16 node scores; K loop of 32
// steps (K=4 each) covers DIM=128. fp32 WMMA keeps full accuracy at zero
// extra bandwidth (X is fp32 in HBM; kernel is bandwidth-bound anyway).
// ---------------------------------------------------------------------------
__global__ __launch_bounds__(256) void gat_scores_wmma(
    const float* __restrict__ x,     // [NNODES][DIM]
    const float* __restrict__ w,     // [2*DIM]  (w_src then w_dst)
    float* __restrict__ s_src,       // [NNODES]
    float* __restrict__ s_dst)       // [NNODES]
{
    const int lane     = threadIdx.x & 31;
    const int waveId   = blockIdx.x * 8 + (threadIdx.x >> 5);
    const int nodeBase = waveId * 16;          // NNODES % 16 == 0 -> full waves
    if (nodeBase >= NNODES) return;            // wave-uniform exit (EXEC stays all-1s)

    const int hi  = lane >> 4;                 // 0: lanes 0-15, 1: lanes 16-31
    const int m   = lane & 15;                 // A row (M) / B-D column (N)
    const int klo = hi * 2;                    // K offset inside each 16x4 step

    // A fragment source: row (nodeBase+m), columns k0+klo .. k0+klo+1
    const float* arow = x + (size_t)(nodeBase + m) * DIM + klo;
    // B fragment source: column N<2 -> w[N*DIM + k], else zero (branchless)
    const bool   bact = (m < 2);
    const float* wcol = w + (bact ? m : 0) * DIM + klo;

    v8f acc = {};
    #pragma unroll
    for (int k0 = 0; k0 < DIM; k0 += 4) {
        // X is read exactly once -> stream it (non-temporal), don't pollute L2
        v2f a = __builtin_nontemporal_load((const v2f*)(arow + k0));
        v2f braw = *(const v2f*)(wcol + k0);   // tiny, hot in L0
        v2f b;
        b.x = bact ? braw.x : 0.0f;            // v_cndmask: no EXEC divergence
        b.y = bact ? braw.y : 0.0f;
        // D = A(16x4) * B(4x16) + C ; fp32, round-to-nearest-even
        acc = __builtin_amdgcn_wmma_f32_16x16x4_f32(
            /*neg_a=*/false, a, /*neg_b=*/false, b,
            /*c_mod=*/(short)0, acc, /*reuse_a=*/false, /*reuse_b=*/false);
    }

    // D layout: column N = lane&15; VGPR v holds row (v + 8*hi).
    // Column 0 = score_src, column 1 = score_dst.
    if (m < 2) {
        float* out = (m == 0) ? s_src : s_dst;
        #pragma unroll
        for (int v = 0; v < 8; ++v)
            out[nodeBase + v + hi * 8] = acc[v];
    }
}

// ---------------------------------------------------------------------------
// Phase 2: per-edge sigmoid(score_src[src] + score_dst[dst]) with clamp.
// 4 edges/thread; 32B vector index loads; 800KB score table gathers hit L2.
// ---------------------------------------------------------------------------
__global__ __launch_bounds__(256) void gat_edge_sigmoid(
    const long long* __restrict__ ei,   // [2][NEDGES] int64
    const float* __restrict__ s_src,
    const float* __restrict__ s_dst,
    float* __restrict__ att)            // [NEDGES]
{
    const int e0 = (blockIdx.x * 256 + threadIdx.x) * 4;
    if (e0 >= NEDGES) return;

    v4i64 si = *(const v4i64*)(ei + e0);           // src indices (row 0)
    v4i64 di = *(const v4i64*)(ei + NEDGES + e0);  // dst indices (row 1)

    v4f r;
    #pragma unroll
    for (int j = 0; j < 4; ++j) {
        float z = s_src[(int)si[j]] + s_dst[(int)di[j]];
        float p = 1.0f / (1.0f + __expf(-z));      // v_exp_f32 fast path
        p = fminf(fmaxf(p, 1e-5f), 0.99999f);
        r[j] = p;
    }
    // written once, never read again -> non-temporal b128 store
    __builtin_nontemporal_store(r, (v4f*)(att + e0));
}

// ---------------------------------------------------------------------------
extern "C" void kernel_launch(void* const* d_in, const int* in_sizes, int n_in,
                              void* d_out, int out_size, void* d_ws, size_t ws_size,
                              hipStream_t stream)
{
    const float*     x  = (const float*)d_in[0];      // (100000,128) fp32
    const long long* ei = (const long long*)d_in[1];  // (2,1600000) int64
    const float*     w  = (const float*)d_in[2];      // (256,) fp32

    float* s_src = (float*)d_ws;                      // 400 KB
    float* s_dst = s_src + NNODES;                    // 400 KB
    float* att   = (float*)d_out;                     // 6.4 MB

    // Phase 1: 6250 waves (16 nodes each), 8 waves/block.
    const int waves   = (NNODES + 15) / 16;           // 6250
    const int blocks1 = (waves + 7) / 8;              // 782
    gat_scores_wmma<<<blocks1, 256, 0, stream>>>(x, w, s_src, s_dst);

    // Phase 2: 4 edges per thread.
    const int threads2 = NEDGES / 4;                  // 400000
    const int blocks2  = (threads2 + 255) / 256;      // 1563
    gat_edge_sigmoid<<<blocks2, 256, 0, stream>>>(ei, s_src, s_dst, att);
}